// GripNet_66340064854089
// MI455X (gfx1250) — compile-verified
//
#include <hip/hip_runtime.h>

typedef __attribute__((ext_vector_type(2))) float v2f;
typedef __attribute__((ext_vector_type(8))) float v8f;

#define NN1 100000
#define NN2 100000

// ---------------------------------------------------------------------------
// FP32 WMMA GEMM with LDS-staged A tile:
//   out[M x Nout] = X[M x K] @ Wm[K x Nout]
// Block = (32, Nout/16): each wave computes one 16x16 n-tile of the same
// 16-row m-block; the 16xK X tile is loaded once (coalesced b128) into LDS
// and shared by all waves. WMMA operands:
//  A 16x4:  lane&15 = M row; lane>>4 selects K pair {0,1}/{2,3}; 2 VGPRs
//  B 4x16:  lane&15 = N col; lane>>4 selects K pair;             2 VGPRs
//  C/D 16x16: vgpr r -> M = r + 8*(lane>>4), N = lane&15
// ---------------------------------------------------------------------------
__global__ __launch_bounds__(128) void wmma_gemm_f32(const float* __restrict__ X,
                                                     const float* __restrict__ Wm,
                                                     float* __restrict__ out,
                                                     int K, int Nout) {
  extern __shared__ float tile[];  // 16 * K floats
  const int lane = threadIdx.x;        // 0..31
  const int m0 = blockIdx.x << 4;

  // cooperative, coalesced load of the 16xK A tile into LDS (float4 chunks)
  const int nth = blockDim.x * blockDim.y;
  const int tid = threadIdx.y * 32 + lane;
  const int total4 = (16 * K) >> 2;
  for (int i = tid; i < total4; i += nth) {
    const int r = (i << 2) / K;
    const int c = (i << 2) % K;
    *(float4*)&tile[r * K + c] = *(const float4*)&X[(m0 + r) * K + c];
  }
  __syncthreads();

  const int half = lane >> 4;
  const int l15 = lane & 15;
  const int n0 = (int)threadIdx.y << 4;
  const int col = n0 + l15;

  v8f acc = {};
  for (int k = 0; k < K; k += 4) {
    const int ka = k + half * 2;
    v2f a = *(const v2f*)&tile[l15 * K + ka];   // ds_load_b64
    v2f b;
    b.x = Wm[ka * Nout + col];
    b.y = Wm[(ka + 1) * Nout + col];
    acc = __builtin_amdgcn_wmma_f32_16x16x4_f32(false, a, false, b, (short)0,
                                                acc, false, false);
  }
  const int orow = m0 + half * 8;
#pragma unroll
  for (int r = 0; r < 8; ++r)
    out[(orow + r) * Nout + col] = acc[r];
}

// ---------------------------------------------------------------------------
// Elementwise / scatter kernels (F is always a power of two: pass log2(F))
// ---------------------------------------------------------------------------
__global__ __launch_bounds__(256) void fill_f32(float* p, float v, long long n) {
  long long i = (long long)blockIdx.x * blockDim.x + threadIdx.x;
  if (i < n) p[i] = v;
}

__global__ __launch_bounds__(256) void count_deg(const int* __restrict__ row,
                                                 float* __restrict__ deg, int E) {
  int e = blockIdx.x * blockDim.x + threadIdx.x;
  if (e >= E) return;
  __hip_atomic_fetch_add(&deg[row[e]], 1.0f, __ATOMIC_RELAXED,
                         __HIP_MEMORY_SCOPE_AGENT);
}

__global__ __launch_bounds__(256) void rsqrt_plus1(const float* __restrict__ deg,
                                                   float* __restrict__ dinv, int n) {
  int i = blockIdx.x * blockDim.x + threadIdx.x;
  if (i < n) dinv[i] = rsqrtf(deg[i] + 1.0f);
}

// agg[c*F+f] += hw[r*F+f] * dinv[r]*dinv[c]   (tid = e*F + f, coalesced in f)
__global__ __launch_bounds__(256) void gcn_scatter(const float* __restrict__ hw,
    const float* __restrict__ dinv, const int* __restrict__ row,
    const int* __restrict__ col, float* __restrict__ agg, long long total,
    int fsh) {
  long long t = (long long)blockIdx.x * blockDim.x + threadIdx.x;
  if (t >= total) return;
  const int F = 1 << fsh;
  int e = (int)(t >> fsh);
  int f = (int)t & (F - 1);
  int r = row[e], c = col[e];
  float nrm = dinv[r] * dinv[c];
  __hip_atomic_fetch_add(&agg[c * F + f], hw[r * F + f] * nrm,
                         __ATOMIC_RELAXED, __HIP_MEMORY_SCOPE_AGENT);
}

// out = relu(agg + hw*dinv^2 + bias)
__global__ __launch_bounds__(256) void gcn_finish(const float* __restrict__ agg,
    const float* __restrict__ hw, const float* __restrict__ dinv,
    const float* __restrict__ bias, float* __restrict__ out, int n, int fsh) {
  int i = blockIdx.x * blockDim.x + threadIdx.x;
  const int F = 1 << fsh;
  if (i >= (n << fsh)) return;
  int node = i >> fsh, f = i & (F - 1);
  float di = dinv[node];
  float v = agg[i] + hw[i] * di * di + bias[f];
  out[i] = v > 0.0f ? v : 0.0f;
}

// ext[c*F+f] += hx[r*F+f] * norm[r]
__global__ __launch_bounds__(256) void ext_scatter(const float* __restrict__ hx,
    const float* __restrict__ norm, const int* __restrict__ row,
    const int* __restrict__ col, float* __restrict__ out, long long total,
    int fsh) {
  long long t = (long long)blockIdx.x * blockDim.x + threadIdx.x;
  if (t >= total) return;
  const int F = 1 << fsh;
  int e = (int)(t >> fsh);
  int f = (int)t & (F - 1);
  int r = row[e], c = col[e];
  __hip_atomic_fetch_add(&out[c * F + f], hx[r * F + f] * norm[r],
                         __ATOMIC_RELAXED, __HIP_MEMORY_SCOPE_AGENT);
}

__global__ __launch_bounds__(256) void bias_relu(float* __restrict__ x,
    const float* __restrict__ bias, int n, int fsh) {
  int i = blockIdx.x * blockDim.x + threadIdx.x;
  if (i >= (n << fsh)) return;
  float v = x[i] + bias[i & ((1 << fsh) - 1)];
  x[i] = v > 0.0f ? v : 0.0f;
}

// h = a + b ; out[node*stride + off + f] = h  (concat slot 0: raw h2)
__global__ __launch_bounds__(256) void add_store(const float* __restrict__ a,
    const float* __restrict__ b, float* __restrict__ h, float* __restrict__ out,
    int n, int fsh, int stride, int off) {
  int i = blockIdx.x * blockDim.x + threadIdx.x;
  if (i >= (n << fsh)) return;
  int node = i >> fsh, f = i & ((1 << fsh) - 1);
  float v = a[i] + b[i];
  h[i] = v;
  out[node * stride + off + f] = v;
}

// Wr[r, :, :] = sum_b comp[r,b] * basis[b, :, :]   (DF = din*dout)
__global__ __launch_bounds__(256) void make_wr(const float* __restrict__ comp,
    const float* __restrict__ basis, float* __restrict__ Wr, int R, int NB, int DF) {
  int i = blockIdx.x * blockDim.x + threadIdx.x;
  if (i >= R * DF) return;
  int r = i / DF, j = i % DF;
  float s = 0.0f;
  for (int b = 0; b < NB; ++b) s += comp[r * NB + b] * basis[b * DF + j];
  Wr[i] = s;
}

__global__ __launch_bounds__(256) void count_deg_rel(const int* __restrict__ col,
    const int* __restrict__ et, float* __restrict__ degr, int E, int n) {
  int e = blockIdx.x * blockDim.x + threadIdx.x;
  if (e >= E) return;
  __hip_atomic_fetch_add(&degr[et[e] * n + col[e]], 1.0f, __ATOMIC_RELAXED,
                         __HIP_MEMORY_SCOPE_AGENT);
}

// agg[c*F+f] += hr[(t*n+r)*F+f] / max(degr[t*n+c], 1)
__global__ __launch_bounds__(256) void rgcn_scatter(const float* __restrict__ hr,
    const float* __restrict__ degr, const int* __restrict__ row,
    const int* __restrict__ col, const int* __restrict__ et,
    float* __restrict__ agg, long long total, int fsh, int n) {
  long long tt = (long long)blockIdx.x * blockDim.x + threadIdx.x;
  if (tt >= total) return;
  const int F = 1 << fsh;
  int e = (int)(tt >> fsh);
  int f = (int)tt & (F - 1);
  int r = row[e], c = col[e], t = et[e];
  float d = degr[t * n + c];
  float inv = 1.0f / (d > 1.0f ? d : 1.0f);
  __hip_atomic_fetch_add(&agg[c * F + f], hr[(t * n + r) * F + f] * inv,
                         __ATOMIC_RELAXED, __HIP_MEMORY_SCOPE_AGENT);
}

// v = relu(agg + xroot + bias); h=v contiguous; out[node*stride+off+f]=v
__global__ __launch_bounds__(256) void rgcn_finish(const float* __restrict__ agg,
    const float* __restrict__ xroot, const float* __restrict__ bias,
    float* __restrict__ h, float* __restrict__ out, int n, int fsh,
    int stride, int off) {
  int i = blockIdx.x * blockDim.x + threadIdx.x;
  if (i >= (n << fsh)) return;
  int node = i >> fsh, f = i & ((1 << fsh) - 1);
  float v = agg[i] + xroot[i] + bias[f];
  v = v > 0.0f ? v : 0.0f;
  h[i] = v;
  out[node * stride + off + f] = v;
}

// ---------------------------------------------------------------------------
static inline int cdiv(long long a, int b) { return (int)((a + b - 1) / b); }

static inline void launch_gemm(const float* X, const float* Wm, float* out,
                               int M, int K, int Nout, hipStream_t stream) {
  dim3 grid(M / 16, 1);
  dim3 block(32, Nout / 16);
  size_t shmem = (size_t)16 * K * sizeof(float);
  wmma_gemm_f32<<<grid, block, shmem, stream>>>(X, Wm, out, K, Nout);
}

extern "C" void kernel_launch(void* const* d_in, const int* in_sizes, int n_in,
                              void* d_out, int out_size, void* d_ws, size_t ws_size,
                              hipStream_t stream) {
  const float* x1    = (const float*)d_in[0];
  const float* x2    = (const float*)d_in[1];
  const int*   ei1   = (const int*)d_in[2];
  const int*   ei12  = (const int*)d_in[3];
  const int*   ei2   = (const int*)d_in[4];
  const int*   et2   = (const int*)d_in[5];
  const float* emb1  = (const float*)d_in[6];
  const float* w_g1  = (const float*)d_in[7];
  const float* b_g1  = (const float*)d_in[8];
  const float* w_g2  = (const float*)d_in[9];
  const float* b_g2  = (const float*)d_in[10];
  const float* w_ext = (const float*)d_in[11];
  const float* b_ext = (const float*)d_in[12];
  const float* emb2  = (const float*)d_in[13];
  const float* basis1= (const float*)d_in[14];
  const float* comp1 = (const float*)d_in[15];
  const float* root1 = (const float*)d_in[16];
  const float* bias1 = (const float*)d_in[17];
  const float* basis2= (const float*)d_in[18];
  const float* comp2 = (const float*)d_in[19];
  const float* root2 = (const float*)d_in[20];
  const float* bias2 = (const float*)d_in[21];

  const int E1  = in_sizes[2] / 2;
  const int E12 = in_sizes[3] / 2;
  const int E2  = in_sizes[4] / 2;
  const int* r1  = ei1;           const int* c1  = ei1  + E1;
  const int* r12 = ei12;          const int* c12 = ei12 + E12;
  const int* r2  = ei2;           const int* c2  = ei2  + E2;

  float* out = (float*)d_out;
  float* WS = (float*)d_ws;
  size_t off = 0;
  auto alloc = [&](size_t n) { float* p = WS + off; off += n; return p; };

  float* A   = alloc((size_t)NN1 * 64); // h1a; later (with Bb) hr [3*N2*32]
  float* Bb  = alloc((size_t)NN1 * 32); // hw1/hw2/hx; later tail of hr
  float* Cb  = alloc((size_t)NN1 * 32); // GCN agg; later RGCN agg
  float* Eb  = alloc((size_t)NN1 * 32); // h1b; later xroot
  float* Fb  = alloc((size_t)NN1 * 32); // h1c; later h2b
  float* Gb  = alloc((size_t)NN2 * 32); // ext; later scratch h2c
  float* Hb  = alloc((size_t)NN2 * 32); // h2
  float* Dg  = alloc(100000);           // deg
  float* Di  = alloc(100000);           // dinv / norm
  float* DR  = alloc((size_t)3 * NN2);  // RGCN per-(rel,dst) degree
  float* WrB = alloc(3 * 32 * 32);      // basis-combined weights
  float* HR  = A;                       // 3*N2*32 spans A+Bb

  const dim3 B256(256);

  // ---- SV1: h1a = x1 @ emb1 -------------------------------------------------
  launch_gemm(x1, emb1, A, NN1, 64, 64, stream);

  // degrees + dinv for graph1
  fill_f32<<<cdiv(NN1, 256), B256, 0, stream>>>(Dg, 0.0f, NN1);
  count_deg<<<cdiv(E1, 256), B256, 0, stream>>>(r1, Dg, E1);
  rsqrt_plus1<<<cdiv(NN1, 256), B256, 0, stream>>>(Dg, Di, NN1);

  // GCN layer 1: 64 -> 32
  launch_gemm(A, w_g1, Bb, NN1, 64, 32, stream);
  fill_f32<<<cdiv((long long)NN1 * 32, 256), B256, 0, stream>>>(Cb, 0.0f, (long long)NN1 * 32);
  gcn_scatter<<<cdiv((long long)E1 * 32, 256), B256, 0, stream>>>(Bb, Di, r1, c1, Cb,
                                                                  (long long)E1 * 32, 5);
  gcn_finish<<<cdiv((long long)NN1 * 32, 256), B256, 0, stream>>>(Cb, Bb, Di, b_g1, Eb, NN1, 5);

  // GCN layer 2: 32 -> 32
  launch_gemm(Eb, w_g2, Bb, NN1, 32, 32, stream);
  fill_f32<<<cdiv((long long)NN1 * 32, 256), B256, 0, stream>>>(Cb, 0.0f, (long long)NN1 * 32);
  gcn_scatter<<<cdiv((long long)E1 * 32, 256), B256, 0, stream>>>(Bb, Di, r1, c1, Cb,
                                                                  (long long)E1 * 32, 5);
  gcn_finish<<<cdiv((long long)NN1 * 32, 256), B256, 0, stream>>>(Cb, Bb, Di, b_g2, Fb, NN1, 5);

  // ---- External SV1 -> SV2 --------------------------------------------------
  launch_gemm(Fb, w_ext, Bb, NN1, 32, 32, stream);
  fill_f32<<<cdiv(NN1, 256), B256, 0, stream>>>(Dg, 0.0f, NN1);
  count_deg<<<cdiv(E12, 256), B256, 0, stream>>>(r12, Dg, E12);
  rsqrt_plus1<<<cdiv(NN1, 256), B256, 0, stream>>>(Dg, Di, NN1);
  fill_f32<<<cdiv((long long)NN2 * 32, 256), B256, 0, stream>>>(Gb, 0.0f, (long long)NN2 * 32);
  ext_scatter<<<cdiv((long long)E12 * 32, 256), B256, 0, stream>>>(Bb, Di, r12, c12, Gb,
                                                                   (long long)E12 * 32, 5);
  bias_relu<<<cdiv((long long)NN2 * 32, 256), B256, 0, stream>>>(Gb, b_ext, NN2, 5);

  // ---- SV2: h2 = x2 @ emb2 + ext ; concat slot 0 ---------------------------
  launch_gemm(x2, emb2, Hb, NN2, 64, 32, stream);
  add_store<<<cdiv((long long)NN2 * 32, 256), B256, 0, stream>>>(Hb, Gb, Hb, out,
                                                                 NN2, 5, 80, 0);

  // per-(relation,dst) degrees (shared by both RGCN layers)
  fill_f32<<<cdiv((long long)3 * NN2, 256), B256, 0, stream>>>(DR, 0.0f, (long long)3 * NN2);
  count_deg_rel<<<cdiv(E2, 256), B256, 0, stream>>>(c2, et2, DR, E2, NN2);

  // ---- RGCN layer 1: 32 -> 32 ----------------------------------------------
  make_wr<<<cdiv(3 * 32 * 32, 256), B256, 0, stream>>>(comp1, basis1, WrB, 3, 4, 32 * 32);
  for (int r = 0; r < 3; ++r)
    launch_gemm(Hb, WrB + (size_t)r * 32 * 32, HR + (size_t)r * NN2 * 32, NN2, 32, 32, stream);
  fill_f32<<<cdiv((long long)NN2 * 32, 256), B256, 0, stream>>>(Cb, 0.0f, (long long)NN2 * 32);
  rgcn_scatter<<<cdiv((long long)E2 * 32, 256), B256, 0, stream>>>(HR, DR, r2, c2, et2, Cb,
                                                                   (long long)E2 * 32, 5, NN2);
  launch_gemm(Hb, root1, Eb, NN2, 32, 32, stream);
  rgcn_finish<<<cdiv((long long)NN2 * 32, 256), B256, 0, stream>>>(Cb, Eb, bias1, Fb, out,
                                                                   NN2, 5, 80, 32);

  // ---- RGCN layer 2: 32 -> 16 ----------------------------------------------
  make_wr<<<cdiv(3 * 32 * 16, 256), B256, 0, stream>>>(comp2, basis2, WrB, 3, 4, 32 * 16);
  for (int r = 0; r < 3; ++r)
    launch_gemm(Fb, WrB + (size_t)r * 32 * 16, HR + (size_t)r * NN2 * 16, NN2, 32, 16, stream);
  fill_f32<<<cdiv((long long)NN2 * 16, 256), B256, 0, stream>>>(Cb, 0.0f, (long long)NN2 * 16);
  rgcn_scatter<<<cdiv((long long)E2 * 16, 256), B256, 0, stream>>>(HR, DR, r2, c2, et2, Cb,
                                                                   (long long)E2 * 16, 4, NN2);
  launch_gemm(Fb, root2, Eb, NN2, 32, 16, stream);
  rgcn_finish<<<cdiv((long long)NN2 * 16, 256), B256, 0, stream>>>(Cb, Eb, bias2, Gb, out,
                                                                   NN2, 4, 80, 64);
}